// Type12_41712722379509
// MI455X (gfx1250) — compile-verified
//
#include <hip/hip_runtime.h>
#include <hip/hip_bf16.h>
#include <math.h>
#include <stdint.h>

// ---------------------------------------------------------------------------
// CDNA5 (gfx1250) fused GCN pipeline.
//   H0 = LN(adj0 @ (x@W0) + b0)
//   H1 = leaky(LN(adj1 @ (H0@W1) + b1))
//   H2 = leaky(LN(adj2 @ (H1@W2) + b2))
//   out = log_softmax(H2 @ Wl + bl)
// Big GEMMs use V_WMMA_F32_16X16X4_F32; adj tiles streamed HBM->LDS with
// async loads (ASYNCcnt) double-buffered against WMMA compute.
// ---------------------------------------------------------------------------

typedef float v2f __attribute__((ext_vector_type(2)));
typedef float v8f __attribute__((ext_vector_type(8)));

__device__ __forceinline__ void async_tile_load_b128(unsigned ldsOff, const float* g) {
    // GLOBAL_LOAD_ASYNC_TO_LDS_B128: VDST = LDS byte offset VGPR, VADDR = 64-bit addr.
    asm volatile("global_load_async_to_lds_b128 %0, %1, off"
                 :: "v"(ldsOff), "v"(g) : "memory");
}

__device__ __forceinline__ void wait_async0() {
#if __has_builtin(__builtin_amdgcn_s_wait_asynccnt)
    __builtin_amdgcn_s_wait_asynccnt(0);
#else
    asm volatile("s_wait_asynccnt 0x0" ::: "memory");
#endif
}

// ---------------------------------------------------------------------------
// Small GEMM: out[N x Hdim] = Hin[N x K] @ W[K x Hdim]   (no bias; bias is
// folded into the gcn kernel). One wave computes one 16x16 tile.
// ---------------------------------------------------------------------------
__global__ __launch_bounds__(32) void xw_kernel(const float* __restrict__ Hin,
                                                const float* __restrict__ W,
                                                float* __restrict__ out,
                                                int N, int K, int Hdim) {
    const int lane = threadIdx.x & 31;
    const int m    = lane & 15;
    const int hi   = lane >> 4;      // 0: lanes 0-15, 1: lanes 16-31
    const int sel  = hi * 2;         // A/B K-offset per ISA layout
    const int row  = blockIdx.x * 16 + m;
    const int col  = blockIdx.y * 16 + m;

    v8f acc = {};
    for (int k = 0; k < K; k += 4) {
        v2f a, b;
        a.x = Hin[(size_t)row * K + k + sel];
        a.y = Hin[(size_t)row * K + k + sel + 1];
        b.x = W[(size_t)(k + sel)     * Hdim + col];
        b.y = W[(size_t)(k + sel + 1) * Hdim + col];
        acc = __builtin_amdgcn_wmma_f32_16x16x4_f32(false, a, false, b,
                                                    (short)0, acc, false, false);
    }
#pragma unroll
    for (int v = 0; v < 8; ++v) {
        int r = blockIdx.x * 16 + v + hi * 8;   // C/D layout: VGPR v -> row v / v+8
        out[(size_t)r * Hdim + col] = acc[v];
    }
}

// ---------------------------------------------------------------------------
// Fused: out = act(LayerNorm(adj @ XW + bias) * gamma + beta)
// HDIM in {128, 64, 32}. 8 waves/block arranged as (8/CT row-tiles) x (CT col
// tiles); adj strip of ROWS rows streamed through double-buffered LDS.
// ---------------------------------------------------------------------------
template <int HDIM>
__global__ __launch_bounds__(256) void gcn_ln_kernel(
        const float* __restrict__ adj, const float* __restrict__ XW,
        const float* __restrict__ bias, const float* __restrict__ gamma,
        const float* __restrict__ beta, float* __restrict__ out,
        int N, int leaky) {
    constexpr int KC   = 64;                     // adj K-chunk per stage
    constexpr int CT   = HDIM / 16;              // column tiles (waves across H)
    constexpr int RT   = 8 / CT;                 // row tiles
    constexpr int ROWS = 16 * RT;                // adj rows per workgroup
    constexpr int LPT  = (ROWS * KC / 4) / 256;  // b128 async loads per thread
    constexpr int LSTR = KC + 4;                 // padded LDS row stride (floats)

    __shared__ float tile[2][ROWS][LSTR];
    __shared__ float red[2][128];                // CT*ROWS == 128 always

    const int tid  = threadIdx.x;
    const int lane = tid & 31;
    const int wave = tid >> 5;
    const int rt   = wave / CT;
    const int ct   = wave % CT;
    const int m    = lane & 15;
    const int hi   = lane >> 4;
    const int sel  = hi * 2;
    const int col  = ct * 16 + m;
    const size_t rowblock = (size_t)blockIdx.x * ROWS;

    auto issue = [&](int kbase, int buf) {
#pragma unroll
        for (int i = 0; i < LPT; ++i) {
            int e  = tid + i * 256;
            int r  = e / (KC / 4);
            int c4 = (e % (KC / 4)) * 4;
            unsigned ldsOff = (unsigned)(size_t)(&tile[buf][r][c4]);
            const float* g = adj + (rowblock + r) * (size_t)N + kbase + c4;
            async_tile_load_b128(ldsOff, g);
        }
    };

    v8f acc = {};
    issue(0, 0);
    const int NCH = N / KC;
    for (int ch = 0; ch < NCH; ++ch) {
        const int buf = ch & 1;
        wait_async0();            // this wave's chunk `ch` is in LDS
        __syncthreads();          // everyone's chunk arrived; prev compute done
        if (ch + 1 < NCH) issue((ch + 1) * KC, buf ^ 1);  // overlap next chunk

        const float* bp = XW + (size_t)(ch * KC) * HDIM + col;
#pragma unroll
        for (int ks = 0; ks < KC; ks += 4) {
            v2f a, b;
            const float* ap = &tile[buf][rt * 16 + m][ks + sel];
            a.x = ap[0];
            a.y = ap[1];
            b.x = bp[(size_t)(ks + sel)     * HDIM];
            b.y = bp[(size_t)(ks + sel + 1) * HDIM];
            acc = __builtin_amdgcn_wmma_f32_16x16x4_f32(false, a, false, b,
                                                        (short)0, acc, false, false);
        }
    }

    // --- bias (pre-LN, per reference: adj@XW + b), per-lane column-uniform ---
    const float bv = bias[col];
#pragma unroll
    for (int v = 0; v < 8; ++v) acc[v] += bv;

    // --- LayerNorm over H: intra-tile reduce across 16 lanes of each half ---
    float s1[8], s2[8];
#pragma unroll
    for (int v = 0; v < 8; ++v) {
        float x = acc[v];
        float y = x * x;
#pragma unroll
        for (int msk = 8; msk >= 1; msk >>= 1) {
            x += __shfl_xor(x, msk, 32);
            y += __shfl_xor(y, msk, 32);
        }
        s1[v] = x;
        s2[v] = y;
    }
    if (m == 0) {
#pragma unroll
        for (int v = 0; v < 8; ++v) {
            int r = rt * 16 + v + hi * 8;
            red[0][ct * ROWS + r] = s1[v];
            red[1][ct * ROWS + r] = s2[v];
        }
    }
    __syncthreads();

    const float gv = gamma[col];
    const float bev = beta[col];
    const float inv = 1.0f / (float)HDIM;
#pragma unroll
    for (int v = 0; v < 8; ++v) {
        int r = rt * 16 + v + hi * 8;
        float su = 0.f, sq = 0.f;
#pragma unroll
        for (int c = 0; c < CT; ++c) {
            su += red[0][c * ROWS + r];
            sq += red[1][c * ROWS + r];
        }
        float mu  = su * inv;
        float var = sq * inv - mu * mu;
        float rs  = rsqrtf(var + 1e-5f);
        float yv  = (acc[v] - mu) * rs * gv + bev;
        if (leaky) yv = (yv > 0.f) ? yv : 0.01f * yv;
        out[(rowblock + r) * HDIM + col] = yv;
    }
}

// ---------------------------------------------------------------------------
// logits = H2[N x 32] @ Wl[32 x 20] + bl, then row-wise log_softmax.
// One wave per row; lane j (<20) owns one class.
// ---------------------------------------------------------------------------
__global__ __launch_bounds__(256) void logits_kernel(const float* __restrict__ H2,
                                                     const float* __restrict__ Wl,
                                                     const float* __restrict__ bl,
                                                     float* __restrict__ out,
                                                     int N) {
    const int lane = threadIdx.x & 31;
    const int wave = threadIdx.x >> 5;
    const int row  = blockIdx.x * 8 + wave;
    constexpr int K = 32, C = 20;

    float logit = -INFINITY;
    if (lane < C) {
        float a = bl[lane];
        for (int k = 0; k < K; ++k)
            a += H2[(size_t)row * K + k] * Wl[k * C + lane];
        logit = a;
    }
    float mx = logit;
#pragma unroll
    for (int msk = 16; msk >= 1; msk >>= 1)
        mx = fmaxf(mx, __shfl_xor(mx, msk, 32));
    float e = (lane < C) ? expf(logit - mx) : 0.f;
    float s = e;
#pragma unroll
    for (int msk = 16; msk >= 1; msk >>= 1)
        s += __shfl_xor(s, msk, 32);
    if (lane < C)
        out[(size_t)row * C + lane] = (logit - mx) - logf(s);
}

// ---------------------------------------------------------------------------
extern "C" void kernel_launch(void* const* d_in, const int* in_sizes, int n_in,
                              void* d_out, int out_size, void* d_ws, size_t ws_size,
                              hipStream_t stream) {
    const float* x    = (const float*)d_in[0];
    const float* adj0 = (const float*)d_in[1];
    const float* adj1 = (const float*)d_in[2];
    const float* adj2 = (const float*)d_in[3];
    const float* W0   = (const float*)d_in[4];
    const float* b0   = (const float*)d_in[5];
    const float* W1   = (const float*)d_in[6];
    const float* b1   = (const float*)d_in[7];
    const float* W2   = (const float*)d_in[8];
    const float* b2   = (const float*)d_in[9];
    const float* g0   = (const float*)d_in[10];
    const float* be0  = (const float*)d_in[11];
    const float* g1   = (const float*)d_in[12];
    const float* be1  = (const float*)d_in[13];
    const float* g2   = (const float*)d_in[14];
    const float* be2  = (const float*)d_in[15];
    const float* Wl   = (const float*)d_in[16];
    const float* bl   = (const float*)d_in[17];
    float* out = (float*)d_out;

    const int N = 12288;
    float* ws0 = (float*)d_ws;              // N*128 floats ping
    float* ws1 = ws0 + (size_t)N * 128;     // N*128 floats pong (12.6 MB total)

    // XW0 = x @ W0                       (N x 128)
    xw_kernel<<<dim3(N / 16, 128 / 16), 32, 0, stream>>>(x, W0, ws0, N, 128, 128);
    // H0 = LN(adj0 @ XW0 + b0)           (N x 128)
    gcn_ln_kernel<128><<<N / 16, 256, 0, stream>>>(adj0, ws0, b0, g0, be0, ws1, N, 0);
    // XW1 = H0 @ W1                      (N x 64)
    xw_kernel<<<dim3(N / 16, 64 / 16), 32, 0, stream>>>(ws1, W1, ws0, N, 128, 64);
    // H1 = leaky(LN(adj1 @ XW1 + b1))    (N x 64)
    gcn_ln_kernel<64><<<N / 32, 256, 0, stream>>>(adj1, ws0, b1, g1, be1, ws1, N, 1);
    // XW2 = H1 @ W2                      (N x 32)
    xw_kernel<<<dim3(N / 16, 32 / 16), 32, 0, stream>>>(ws1, W2, ws0, N, 64, 32);
    // H2 = leaky(LN(adj2 @ XW2 + b2))    (N x 32)
    gcn_ln_kernel<32><<<N / 64, 256, 0, stream>>>(adj2, ws0, b2, g2, be2, ws1, N, 1);
    // out = log_softmax(H2 @ Wl + bl)    (N x 20)
    logits_kernel<<<N / 8, 256, 0, stream>>>(ws1, Wl, bl, out, N);
}